// MambaBlock_78486232367500
// MI455X (gfx1250) — compile-verified
//
#include <hip/hip_runtime.h>
#include <hip/hip_bf16.h>

// ---------------------------------------------------------------------------
// Mamba block for MI455X (gfx1250). FP32 end-to-end using V_WMMA_F32_16X16X4_F32
// for all GEMMs (precision-exact vs fp32 reference), chunked parallel scan for
// the selective-state recurrence, LDS-staged A tiles for the big GEMMs.
// ---------------------------------------------------------------------------

#define B_      2
#define L_      4096
#define NT      8192        // B*L tokens
#define DM      256         // d_model
#define DI      512         // d_inner
#define E2      1024        // 2*d_inner
#define DS      16          // d_state
#define DR      16          // dt_rank
#define NX      48          // dt_rank + 2*d_state
#define NCHUNK  64
#define CLEN    64          // NCHUNK*CLEN == L_

typedef __attribute__((ext_vector_type(2))) float v2f;
typedef __attribute__((ext_vector_type(8))) float v8f;

__device__ __forceinline__ v8f wmma_f32(v2f a, v2f b, v8f c) {
  // D = A(16x4) * B(4x16) + C(16x16), all fp32, wave32.
  return __builtin_amdgcn_wmma_f32_16x16x4_f32(
      /*neg_a=*/false, a, /*neg_b=*/false, b,
      /*c_mod=*/(short)0, c, /*reuse_a=*/false, /*reuse_b=*/false);
}

__device__ __forceinline__ float sigmoid_f(float v) {
  return 1.0f / (1.0f + __expf(-v));
}

// ---------------------------------------------------------------------------
// Weight prep: transposes for coalesced B-operand loads, A = -exp(A_log).
// ---------------------------------------------------------------------------
__global__ void k_prep(const float* __restrict__ W_in, const float* __restrict__ W_x,
                       const float* __restrict__ W_dt, const float* __restrict__ W_out,
                       const float* __restrict__ A_log,
                       float* __restrict__ WtIn, float* __restrict__ WtX,
                       float* __restrict__ WtDt, float* __restrict__ WtOut,
                       float* __restrict__ Aneg) {
  int stride = gridDim.x * blockDim.x;
  int tid = blockIdx.x * blockDim.x + threadIdx.x;
  for (int i = tid; i < E2 * DM; i += stride) { int e = i / DM, d = i % DM; WtIn[d * E2 + e] = W_in[i]; }
  for (int i = tid; i < NX * DI; i += stride) { int e = i / DI, d = i % DI; WtX[d * NX + e] = W_x[i]; }
  for (int i = tid; i < DI * DR; i += stride) { int e = i / DR, r = i % DR; WtDt[r * DI + e] = W_dt[i]; }
  for (int i = tid; i < DM * DI; i += stride) { int e = i / DI, d = i % DI; WtOut[d * DM + e] = W_out[i]; }
  for (int i = tid; i < DI * DS; i += stride) { Aneg[i] = -__expf(A_log[i]); }
}

// ---------------------------------------------------------------------------
// LayerNorm over d_model=256: one wave per token, 8 elems per lane.
// ---------------------------------------------------------------------------
__global__ void k_layernorm(const float* __restrict__ x, const float* __restrict__ g,
                            const float* __restrict__ bia, float* __restrict__ xn) {
  int wid = (blockIdx.x * blockDim.x + threadIdx.x) >> 5;  // token
  int lane = threadIdx.x & 31;
  const float* row = x + (size_t)wid * DM;
  float v[8], s = 0.f, s2 = 0.f;
#pragma unroll
  for (int j = 0; j < 8; ++j) { v[j] = row[lane + 32 * j]; s += v[j]; s2 += v[j] * v[j]; }
#pragma unroll
  for (int off = 16; off > 0; off >>= 1) { s += __shfl_xor(s, off, 32); s2 += __shfl_xor(s2, off, 32); }
  float mu = s * (1.f / DM);
  float var = s2 * (1.f / DM) - mu * mu;
  float rs = rsqrtf(var + 1e-5f);
#pragma unroll
  for (int j = 0; j < 8; ++j) {
    int c = lane + 32 * j;
    xn[(size_t)wid * DM + c] = (v[j] - mu) * rs * g[c] + bia[c];
  }
}

// ---------------------------------------------------------------------------
// GEMM-in: xz[8192,1024] = xn[8192,256] @ WtIn[256,1024].
// Block = 4 waves sharing one LDS A-tile (16x256). Each wave: 16x64 output.
// ---------------------------------------------------------------------------
__global__ void __launch_bounds__(128) k_gemm_in(const float* __restrict__ xn,
                                                 const float* __restrict__ Wt,
                                                 float* __restrict__ xz) {
  __shared__ float As[16 * 256];  // 16KB, contiguous rows of xn
  int mt = blockIdx.x >> 2;       // 512 M tiles
  int cg = blockIdx.x & 3;        // 4 column groups of 256
  int w = threadIdx.x >> 5;
  int lane = threadIdx.x & 31;
  const float* abase = xn + (size_t)mt * 16 * 256;
  for (int i = threadIdx.x; i < 16 * 256; i += 128) As[i] = abase[i];  // coalesced
  __syncthreads();
  int n0 = cg * 256 + w * 64;
  int mrow = lane & 15, hi = lane >> 4;
  v8f acc[4] = {};
  for (int k0 = 0; k0 < 256; k0 += 4) {
    __builtin_prefetch(&Wt[(size_t)(k0 + 8) * E2 + n0 + mrow], 0, 3);
    v2f a;
    a[0] = As[mrow * 256 + k0 + 2 * hi];
    a[1] = As[mrow * 256 + k0 + 2 * hi + 1];
#pragma unroll
    for (int j = 0; j < 4; ++j) {
      v2f bb;
      bb[0] = Wt[(size_t)(k0 + 2 * hi) * E2 + n0 + 16 * j + mrow];
      bb[1] = Wt[(size_t)(k0 + 2 * hi + 1) * E2 + n0 + 16 * j + mrow];
      acc[j] = wmma_f32(a, bb, acc[j]);
    }
  }
#pragma unroll
  for (int j = 0; j < 4; ++j)
#pragma unroll
    for (int r = 0; r < 8; ++r) {
      int m = r + 8 * hi;
      xz[(size_t)(mt * 16 + m) * E2 + n0 + 16 * j + (lane & 15)] = acc[j][r];
    }
}

// ---------------------------------------------------------------------------
// Depthwise causal conv(4) + bias + SiLU on u = xz[:, :512].
// ---------------------------------------------------------------------------
__global__ void k_conv(const float* __restrict__ xz, const float* __restrict__ cw,
                       const float* __restrict__ cb, float* __restrict__ uc) {
  int idx = blockIdx.x * blockDim.x + threadIdx.x;  // NT*DI threads
  int g = idx >> 9, c = idx & (DI - 1);
  int b = g >> 12, t = g & (L_ - 1);
  float acc = cb[c];
#pragma unroll
  for (int j = 0; j < 4; ++j) {
    int tt = t - 3 + j;
    if (tt >= 0) acc += cw[c * 4 + j] * xz[(size_t)(b * L_ + tt) * E2 + c];
  }
  uc[idx] = acc * sigmoid_f(acc);
}

// ---------------------------------------------------------------------------
// GEMM-x: dbc[8192,48] = uc[8192,512] @ WtX[512,48]. One wave per M tile.
// ---------------------------------------------------------------------------
__global__ void __launch_bounds__(128) k_gemm_x(const float* __restrict__ uc,
                                                const float* __restrict__ WtX,
                                                float* __restrict__ dbc) {
  int wid = (blockIdx.x * blockDim.x + threadIdx.x) >> 5;  // 512 M tiles
  int lane = threadIdx.x & 31;
  int mrow = lane & 15, hi = lane >> 4;
  const float* abase = uc + (size_t)wid * 16 * DI;
  v8f acc[3] = {};
  for (int k0 = 0; k0 < DI; k0 += 4) {
    v2f a;
    a[0] = abase[mrow * DI + k0 + 2 * hi];
    a[1] = abase[mrow * DI + k0 + 2 * hi + 1];
#pragma unroll
    for (int j = 0; j < 3; ++j) {
      v2f bb;
      bb[0] = WtX[(size_t)(k0 + 2 * hi) * NX + 16 * j + mrow];
      bb[1] = WtX[(size_t)(k0 + 2 * hi + 1) * NX + 16 * j + mrow];
      acc[j] = wmma_f32(a, bb, acc[j]);
    }
  }
#pragma unroll
  for (int j = 0; j < 3; ++j)
#pragma unroll
    for (int r = 0; r < 8; ++r)
      dbc[(size_t)(wid * 16 + r + 8 * hi) * NX + 16 * j + (lane & 15)] = acc[j][r];
}

// ---------------------------------------------------------------------------
// GEMM-dt + fused softplus: dt[8192,512] = softplus(dbc[:, :16] @ WtDt + b_dt).
// Block = 8 waves cover one M tile x all 512 cols. K = 16 (4 WMMA steps).
// ---------------------------------------------------------------------------
__global__ void __launch_bounds__(256) k_gemm_dt(const float* __restrict__ dbc,
                                                 const float* __restrict__ WtDt,
                                                 const float* __restrict__ b_dt,
                                                 float* __restrict__ dt) {
  int mt = blockIdx.x;
  int w = threadIdx.x >> 5;
  int lane = threadIdx.x & 31;
  int n0 = w * 64;
  int mrow = lane & 15, hi = lane >> 4;
  v8f acc[4] = {};
#pragma unroll
  for (int k0 = 0; k0 < DR; k0 += 4) {
    v2f a;
    a[0] = dbc[(size_t)(mt * 16 + mrow) * NX + k0 + 2 * hi];
    a[1] = dbc[(size_t)(mt * 16 + mrow) * NX + k0 + 2 * hi + 1];
#pragma unroll
    for (int j = 0; j < 4; ++j) {
      v2f bb;
      bb[0] = WtDt[(size_t)(k0 + 2 * hi) * DI + n0 + 16 * j + mrow];
      bb[1] = WtDt[(size_t)(k0 + 2 * hi + 1) * DI + n0 + 16 * j + mrow];
      acc[j] = wmma_f32(a, bb, acc[j]);
    }
  }
#pragma unroll
  for (int j = 0; j < 4; ++j)
#pragma unroll
    for (int r = 0; r < 8; ++r) {
      int col = n0 + 16 * j + (lane & 15);
      float vpre = acc[j][r] + b_dt[col];
      float sp = (vpre > 20.f) ? vpre : __logf(1.f + __expf(vpre));
      dt[(size_t)(mt * 16 + r + 8 * hi) * DI + col] = sp;
    }
}

// ---------------------------------------------------------------------------
// Scan pass A: per (b, channel, chunk) compute chunk summary
//   ap[s] = prod_t dA_t[s],  bc[s] = state reached from h0=0.
// ---------------------------------------------------------------------------
__global__ void k_scan_chunk(const float* __restrict__ dt, const float* __restrict__ uc,
                             const float* __restrict__ dbc, const float* __restrict__ Aneg,
                             float* __restrict__ chA, float* __restrict__ chB) {
  int idx = blockIdx.x * blockDim.x + threadIdx.x;  // B_*NCHUNK*DI
  int c = idx & (DI - 1);
  int chunk = (idx >> 9) & (NCHUNK - 1);
  int b = idx >> 15;
  float As[DS], ap[DS], bc[DS];
#pragma unroll
  for (int s = 0; s < DS; ++s) { As[s] = Aneg[c * DS + s]; ap[s] = 1.f; bc[s] = 0.f; }
  int gbase = b * L_ + chunk * CLEN;
  for (int t = 0; t < CLEN; ++t) {
    int g = gbase + t;
    float dtv = dt[(size_t)g * DI + c];
    float uv = uc[(size_t)g * DI + c];
    float du = dtv * uv;
#pragma unroll
    for (int s = 0; s < DS; ++s) {
      float Bv = dbc[(size_t)g * NX + DR + s];  // uniform across lanes -> broadcast
      float dA = __expf(dtv * As[s]);
      ap[s] *= dA;
      bc[s] = dA * bc[s] + du * Bv;
    }
  }
  size_t o = ((size_t)(b * DI + c) * NCHUNK + chunk) * DS;
#pragma unroll
  for (int q = 0; q < 4; ++q) {
    ((float4*)(chA + o))[q] = make_float4(ap[4 * q], ap[4 * q + 1], ap[4 * q + 2], ap[4 * q + 3]);
    ((float4*)(chB + o))[q] = make_float4(bc[4 * q], bc[4 * q + 1], bc[4 * q + 2], bc[4 * q + 3]);
  }
}

// ---------------------------------------------------------------------------
// Scan pass B: sequential carry over the 64 chunk summaries (per b,c,s).
// ---------------------------------------------------------------------------
__global__ void k_scan_carry(const float* __restrict__ chA, const float* __restrict__ chB,
                             float* __restrict__ hin) {
  int idx = blockIdx.x * blockDim.x + threadIdx.x;  // B_*DI*DS
  int base = idx >> 4;   // b*DI + c
  int s = idx & (DS - 1);
  float h = 0.f;
  for (int chunk = 0; chunk < NCHUNK; ++chunk) {
    size_t o = ((size_t)base * NCHUNK + chunk) * DS + s;
    hin[o] = h;
    h = chA[o] * h + chB[o];
  }
}

// ---------------------------------------------------------------------------
// Scan pass C: replay each chunk from its true h_init, emit
//   y = scan_y + D*u_c, gated by silu(z). ybuf[8192,512].
// ---------------------------------------------------------------------------
__global__ void k_scan_apply(const float* __restrict__ dt, const float* __restrict__ uc,
                             const float* __restrict__ dbc, const float* __restrict__ Aneg,
                             const float* __restrict__ hin, const float* __restrict__ Dp,
                             const float* __restrict__ xz, float* __restrict__ ybuf) {
  int idx = blockIdx.x * blockDim.x + threadIdx.x;
  int c = idx & (DI - 1);
  int chunk = (idx >> 9) & (NCHUNK - 1);
  int b = idx >> 15;
  float As[DS], h[DS];
#pragma unroll
  for (int s = 0; s < DS; ++s) As[s] = Aneg[c * DS + s];
  size_t o = ((size_t)(b * DI + c) * NCHUNK + chunk) * DS;
#pragma unroll
  for (int q = 0; q < 4; ++q) {
    float4 hv = ((const float4*)(hin + o))[q];
    h[4 * q] = hv.x; h[4 * q + 1] = hv.y; h[4 * q + 2] = hv.z; h[4 * q + 3] = hv.w;
  }
  float dpc = Dp[c];
  int gbase = b * L_ + chunk * CLEN;
  for (int t = 0; t < CLEN; ++t) {
    int g = gbase + t;
    float dtv = dt[(size_t)g * DI + c];
    float uv = uc[(size_t)g * DI + c];
    float du = dtv * uv;
    float y = 0.f;
#pragma unroll
    for (int s = 0; s < DS; ++s) {
      float Bv = dbc[(size_t)g * NX + DR + s];
      float Cv = dbc[(size_t)g * NX + DR + DS + s];
      float dA = __expf(dtv * As[s]);
      h[s] = dA * h[s] + du * Bv;
      y += h[s] * Cv;
    }
    float yv = y + dpc * uv;
    float zv = xz[(size_t)g * E2 + DI + c];
    yv *= zv * sigmoid_f(zv);
    ybuf[(size_t)g * DI + c] = yv;
  }
}

// ---------------------------------------------------------------------------
// GEMM-out + residual: out[8192,256] = ybuf[8192,512] @ WtOut[512,256] + x.
// Block = 4 waves share LDS A-tile (16x512 = 32KB); each wave: 16x64.
// ---------------------------------------------------------------------------
__global__ void __launch_bounds__(128) k_gemm_out(const float* __restrict__ ybuf,
                                                  const float* __restrict__ WtOut,
                                                  const float* __restrict__ x,
                                                  float* __restrict__ out) {
  __shared__ float As[16 * 512];  // 32KB
  int mt = blockIdx.x;            // 512 M tiles, N covered fully (4 waves x 64)
  int w = threadIdx.x >> 5;
  int lane = threadIdx.x & 31;
  const float* abase = ybuf + (size_t)mt * 16 * DI;
  for (int i = threadIdx.x; i < 16 * DI; i += 128) As[i] = abase[i];
  __syncthreads();
  int n0 = w * 64;
  int mrow = lane & 15, hi = lane >> 4;
  v8f acc[4] = {};
  for (int k0 = 0; k0 < DI; k0 += 4) {
    __builtin_prefetch(&WtOut[(size_t)(k0 + 8) * DM + n0 + mrow], 0, 3);
    v2f a;
    a[0] = As[mrow * DI + k0 + 2 * hi];
    a[1] = As[mrow * DI + k0 + 2 * hi + 1];
#pragma unroll
    for (int j = 0; j < 4; ++j) {
      v2f bb;
      bb[0] = WtOut[(size_t)(k0 + 2 * hi) * DM + n0 + 16 * j + mrow];
      bb[1] = WtOut[(size_t)(k0 + 2 * hi + 1) * DM + n0 + 16 * j + mrow];
      acc[j] = wmma_f32(a, bb, acc[j]);
    }
  }
#pragma unroll
  for (int j = 0; j < 4; ++j)
#pragma unroll
    for (int r = 0; r < 8; ++r) {
      int m = r + 8 * hi;
      size_t oi = (size_t)(mt * 16 + m) * DM + n0 + 16 * j + (lane & 15);
      out[oi] = acc[j][r] + x[oi];
    }
}

// ---------------------------------------------------------------------------
extern "C" void kernel_launch(void* const* d_in, const int* in_sizes, int n_in,
                              void* d_out, int out_size, void* d_ws, size_t ws_size,
                              hipStream_t stream) {
  const float* x      = (const float*)d_in[0];
  const float* ln_g   = (const float*)d_in[1];
  const float* ln_b   = (const float*)d_in[2];
  const float* W_in   = (const float*)d_in[3];
  const float* conv_w = (const float*)d_in[4];
  const float* conv_b = (const float*)d_in[5];
  const float* W_x    = (const float*)d_in[6];
  const float* W_dt   = (const float*)d_in[7];
  const float* b_dt   = (const float*)d_in[8];
  const float* A_log  = (const float*)d_in[9];
  const float* D_par  = (const float*)d_in[10];
  const float* W_out  = (const float*)d_in[11];
  float* out = (float*)d_out;

  float* ws    = (float*)d_ws;
  float* xn    = ws;                                  // 8192*256
  float* xz    = xn    + (size_t)NT * DM;             // 8192*1024
  float* uc    = xz    + (size_t)NT * E2;             // 8192*512
  float* dbc   = uc    + (size_t)NT * DI;             // 8192*48
  float* dtb   = dbc   + (size_t)NT * NX;             // 8192*512
  float* ybuf  = dtb   + (size_t)NT * DI;             // 8192*512
  float* WtIn  = ybuf  + (size_t)NT * DI;             // 256*1024
  float* WtX   = WtIn  + (size_t)DM * E2;             // 512*48
  float* WtDt  = WtX   + (size_t)DI * NX;             // 16*512
  float* WtOut = WtDt  + (size_t)DR * DI;             // 512*256
  float* Aneg  = WtOut + (size_t)DI * DM;             // 512*16
  float* chA   = Aneg  + (size_t)DI * DS;             // 2*512*64*16
  float* chB   = chA   + (size_t)B_ * DI * NCHUNK * DS;
  float* hin   = chB   + (size_t)B_ * DI * NCHUNK * DS;

  k_prep<<<256, 256, 0, stream>>>(W_in, W_x, W_dt, W_out, A_log, WtIn, WtX, WtDt, WtOut, Aneg);
  k_layernorm<<<NT / 8, 256, 0, stream>>>(x, ln_g, ln_b, xn);
  k_gemm_in<<<512 * 4, 128, 0, stream>>>(xn, WtIn, xz);
  k_conv<<<(NT * DI) / 256, 256, 0, stream>>>(xz, conv_w, conv_b, uc);
  k_gemm_x<<<128, 128, 0, stream>>>(uc, WtX, dbc);
  k_gemm_dt<<<512, 256, 0, stream>>>(dbc, WtDt, b_dt, dtb);
  k_scan_chunk<<<(B_ * DI * NCHUNK) / 256, 256, 0, stream>>>(dtb, uc, dbc, Aneg, chA, chB);
  k_scan_carry<<<(B_ * DI * DS) / 256, 256, 0, stream>>>(chA, chB, hin);
  k_scan_apply<<<(B_ * DI * NCHUNK) / 256, 256, 0, stream>>>(dtb, uc, dbc, Aneg, hin, D_par, xz, ybuf);
  k_gemm_out<<<512, 128, 0, stream>>>(ybuf, WtOut, x, out);
}